// Aposcore_81252191305788
// MI455X (gfx1250) — compile-verified
//
#include <hip/hip_runtime.h>
#include <hip/hip_bf16.h>

typedef __attribute__((ext_vector_type(2))) float v2f;
typedef __attribute__((ext_vector_type(8))) float v8f;

#define B_    16
#define N_L   64
#define N_P   512
#define H_    128
#define G_    10
#define MROWS (B_*N_L*N_P)   /* 524288 */
#define ATOMS 24

#define WAVES   4            /* waves per block, 128 threads */
#define ASTRIDE 132          /* padded LDS row stride (floats) for A / W */
#define CSTRIDE 33           /* padded LDS row stride for C tile */

// ---------------------------------------------------------------------------
// GEMM + epilogue kernel: D(16x32) = A(16x128) * W(128x32), fp32 WMMA.
// W cols: [0..9]=W_pi, [10..19]=W_sigma, [20..29]=W_mu, [30..31]=pad.
// ---------------------------------------------------------------------------
__global__ __launch_bounds__(128) void mdn_head_kernel(
    const float* __restrict__ Interact,
    const float* __restrict__ Wpi,  const float* __restrict__ bpi,
    const float* __restrict__ Wsig, const float* __restrict__ bsig,
    const float* __restrict__ Wmu,  const float* __restrict__ bmu,
    float* __restrict__ out_pi, float* __restrict__ out_sigma,
    float* __restrict__ out_mu)
{
    __shared__ float wB[32 * ASTRIDE];              // column-major: wB[n*132 + k]
    __shared__ float bb[32];
    __shared__ float aL[WAVES * 16 * ASTRIDE];      // per-wave A tiles
    __shared__ float cL[WAVES * 16 * CSTRIDE];      // per-wave D tiles
    __shared__ float pimax[WAVES * 16];
    __shared__ float pisum[WAVES * 16];

    const int tid  = threadIdx.x;
    const int lane = tid & 31;
    const int wave = tid >> 5;

    // ---- pack weights column-major into LDS (zero-pad cols 30..31) ----
    for (int t = tid; t < 32 * H_; t += blockDim.x) {
        const int n = t >> 7;          // 0..31
        const int k = t & 127;         // 0..127
        float w = 0.f;
        if      (n < G_)     w = Wpi [k * G_ + n];
        else if (n < 2*G_)   w = Wsig[k * G_ + (n - G_)];
        else if (n < 3*G_)   w = Wmu [k * G_ + (n - 2*G_)];
        wB[n * ASTRIDE + k] = w;
    }
    if (tid < 32) {
        float b = 0.f;
        if      (tid < G_)     b = bpi [tid];
        else if (tid < 2*G_)   b = bsig[tid - G_];
        else if (tid < 3*G_)   b = bmu [tid - 2*G_];
        bb[tid] = b;
    }

    // ---- stage this wave's 16x128 A tile (coalesced float4 loads) ----
    const long tile = (long)blockIdx.x * WAVES + wave;    // 16-row tile id
    const long row0 = tile * 16;
    const float* Arow = Interact + row0 * H_;
    float* a = aL + wave * (16 * ASTRIDE);
    #pragma unroll 4
    for (int i = 0; i < 16; ++i) {
        const float4 v = ((const float4*)(Arow + (long)i * H_))[lane];
        *(float4*)(a + i * ASTRIDE + lane * 4) = v;       // 16B-aligned ds_store_b128
    }
    __syncthreads();

    // ---- WMMA main loop ----
    // A 16x4 fp32 layout: lane<16 holds A[lane][4kk+0..1] (VGPR0..1),
    //                     lane>=16 holds A[lane-16][4kk+2..3].
    // B 4x16 mirrors with N across lanes -> read from column-major wB.
    const int half = lane >> 4;
    const int lm   = lane & 15;
    v8f c0 = {0.f,0.f,0.f,0.f,0.f,0.f,0.f,0.f};
    v8f c1 = c0;
    #pragma unroll
    for (int kk = 0; kk < 32; ++kk) {
        const int k = 4 * kk + 2 * half;
        const v2f av = *(const v2f*)(a  +  lm       * ASTRIDE + k);
        const v2f b0 = *(const v2f*)(wB +  lm       * ASTRIDE + k);
        const v2f b1 = *(const v2f*)(wB + (lm + 16) * ASTRIDE + k);
        c0 = __builtin_amdgcn_wmma_f32_16x16x4_f32(false, av, false, b0,
                                                   (short)0, c0, false, false);
        c1 = __builtin_amdgcn_wmma_f32_16x16x4_f32(false, av, false, b1,
                                                   (short)0, c1, false, false);
    }

    // ---- bias + spill D tile to LDS (C/D layout: col=lane%16(+16), row=vgpr+8*half)
    float* ct = cL + wave * (16 * CSTRIDE);
    const float bv0 = bb[lm];
    const float bv1 = bb[lm + 16];
    #pragma unroll
    for (int j = 0; j < 8; ++j) {
        const int r = j + 8 * half;
        ct[r * CSTRIDE + lm]      = c0[j] + bv0;
        ct[r * CSTRIDE + lm + 16] = c1[j] + bv1;
    }
    __syncthreads();

    // ---- softmax row statistics (one row per lane, lanes 0..15) ----
    if (lane < 16) {
        const float* rowp = ct + lane * CSTRIDE;
        float m = rowp[0];
        #pragma unroll
        for (int g = 1; g < G_; ++g) m = fmaxf(m, rowp[g]);
        float s = 0.f;
        #pragma unroll
        for (int g = 0; g < G_; ++g) s += __expf(rowp[g] - m);
        pimax[wave * 16 + lane] = m;
        pisum[wave * 16 + lane] = s;
    }
    __syncthreads();

    // ---- epilogues + coalesced stores (160 contiguous floats per stream) ----
    const long obase = row0 * G_;
    for (int idx = lane; idx < 16 * G_; idx += 32) {
        const int r = idx / G_;
        const int g = idx - r * G_;
        const float* rowp = ct + r * CSTRIDE;
        const float m = pimax[wave * 16 + r];
        const float s = pisum[wave * 16 + r];
        out_pi[obase + idx] = __expf(rowp[g] - m) / s;

        float xs = rowp[G_ + g];
        xs = (xs > 0.f) ? xs : 0.01f * xs;                // LeakyReLU(0.01)
        out_sigma[obase + idx] = fmaxf(xs + 1.1f, 1e-6f); // clip(…,1e-6,inf)

        float xm = rowp[2 * G_ + g];
        xm = (xm > 0.f) ? xm : 0.01f * xm;
        out_mu[obase + idx] = xm + 1.0f;
    }
}

// ---------------------------------------------------------------------------
// Min-distance over 24 atoms + mask materialization.
// ---------------------------------------------------------------------------
__global__ __launch_bounds__(256) void dist_kernel(
    const float* __restrict__ pos_l, const float* __restrict__ pos_p,
    const int* __restrict__ imask,
    float* __restrict__ out_dist, float* __restrict__ out_mask)
{
    const int idx = blockIdx.x * 256 + threadIdx.x;
    if (idx >= MROWS) return;
    const int p  = idx & (N_P - 1);
    const int bl = idx >> 9;                 // b*N_l + l
    const int b  = bl >> 6;

    const float* xp = pos_l + (long)bl * 3;
    const float x0 = xp[0], x1 = xp[1], x2 = xp[2];
    const float* Y = pos_p + (long)(b * N_P + p) * (ATOMS * 3);

    float d2min = 3.402823466e38f;
    #pragma unroll 4
    for (int t = 0; t < ATOMS; ++t) {
        const float dx = x0 - Y[t * 3 + 0];
        const float dy = x1 - Y[t * 3 + 1];
        const float dz = x2 - Y[t * 3 + 2];
        d2min = fminf(d2min, dx * dx + dy * dy + dz * dz);
    }
    const float d = sqrtf(d2min);            // d2min >= 0 -> no NaN path needed
    const bool mk = imask[idx] != 0;
    out_dist[idx] = mk ? d : 0.f;
    out_mask[idx] = mk ? 1.f : 0.f;
}

// ---------------------------------------------------------------------------
extern "C" void kernel_launch(void* const* d_in, const int* in_sizes, int n_in,
                              void* d_out, int out_size, void* d_ws, size_t ws_size,
                              hipStream_t stream) {
    const float* pos_l    = (const float*)d_in[0];
    const float* pos_p    = (const float*)d_in[1];
    const float* Interact = (const float*)d_in[2];
    const int*   imask    = (const int*)d_in[3];
    const float* Wpi  = (const float*)d_in[4];
    const float* bpi  = (const float*)d_in[5];
    const float* Wsig = (const float*)d_in[6];
    const float* bsig = (const float*)d_in[7];
    const float* Wmu  = (const float*)d_in[8];
    const float* bmu  = (const float*)d_in[9];

    float* out_pi    = (float*)d_out;
    float* out_sigma = out_pi    + (size_t)MROWS * G_;
    float* out_mu    = out_sigma + (size_t)MROWS * G_;
    float* out_dist  = out_mu    + (size_t)MROWS * G_;
    float* out_mask  = out_dist  + (size_t)MROWS;

    const int tiles  = MROWS / 16;        // 32768
    const int blocks = tiles / WAVES;     // 8192
    mdn_head_kernel<<<dim3(blocks), dim3(128), 0, stream>>>(
        Interact, Wpi, bpi, Wsig, bsig, Wmu, bmu, out_pi, out_sigma, out_mu);

    dist_kernel<<<dim3(MROWS / 256), dim3(256), 0, stream>>>(
        pos_l, pos_p, imask, out_dist, out_mask);
}